// AlignmentEncoder_4896262717839
// MI455X (gfx1250) — compile-verified
//
#include <hip/hip_runtime.h>
#include <hip/hip_bf16.h>
#include <math.h>

typedef __attribute__((ext_vector_type(2))) float v2f;
typedef __attribute__((ext_vector_type(8))) float v8f;

__device__ __forceinline__ v8f wmma4(v2f a, v2f b, v8f c) {
  // D = A(16x4,f32) * B(4x16,f32) + C(16x16,f32)  ->  v_wmma_f32_16x16x4_f32
  return __builtin_amdgcn_wmma_f32_16x16x4_f32(false, a, false, b, (short)0, c,
                                               false, false);
}

// ---------------------------------------------------------------------------
// 1x1 conv as GEMM. One wave computes a 16 x (16*NT) output strip.
// Y[b][m][t] = act( sum_k A[m][k] * X[b][k][t] + bias[m] )
// All operands addressed as SGPR base + 32-bit VGPR element offset so loads
// lower to global_load (saddr+voffset+scale_offset), never FLAT.
// Per K-step: issue all loads first, then the NT WMMAs (clause-friendly).
// ---------------------------------------------------------------------------
template <int NT>
__global__ void __launch_bounds__(32)
conv1x1_gemm_wmma(const float* __restrict__ A, const float* __restrict__ X,
                  const float* __restrict__ bias, float* __restrict__ Y,
                  int M, int K, int T, int relu)
{
  const int lane = threadIdx.x;
  const int row  = lane & 15;
  const int half = lane >> 4;
  const int n0 = blockIdx.x * (16 * NT);
  const int m0 = blockIdx.y << 4;
  const int mrow = m0 + row;

  unsigned bu[NT], tu[NT], offB[NT];
#pragma unroll
  for (int u = 0; u < NT; ++u) {
    const int nc = n0 + 16 * u + row;
    const int b = nc / T;
    const int t = nc - b * T;
    bu[u] = (unsigned)b; tu[u] = (unsigned)t;
    offB[u] = (unsigned)b * K * T + (unsigned)(half << 1) * T + (unsigned)t;
  }
  unsigned offA = (unsigned)mrow * K + (unsigned)(half << 1);
  const unsigned strideT = (unsigned)T;

  v8f acc[NT];
#pragma unroll
  for (int u = 0; u < NT; ++u) acc[u] = {};

  for (int k = 0; k < K; k += 4) {
    __builtin_prefetch(&A[offA + 64], 0, 3);    // global_prefetch_b8 (near)
    const float a0 = A[offA];
    const float a1 = A[offA + 1];
    offA += 4;
    float x0[NT], x1[NT];
#pragma unroll
    for (int u = 0; u < NT; ++u) {
      x0[u] = X[offB[u]];
      x1[u] = X[offB[u] + strideT];
      offB[u] += 4u * strideT;
    }
    const v2f av = { a0, a1 };
#pragma unroll
    for (int u = 0; u < NT; ++u) {
      const v2f bv = { x0[u], x1[u] };
      acc[u] = wmma4(av, bv, acc[u]);
    }
  }

  float bb[8];
#pragma unroll
  for (int i = 0; i < 8; ++i) bb[i] = bias[m0 + i + (half << 3)];
#pragma unroll
  for (int u = 0; u < NT; ++u) {
#pragma unroll
    for (int i = 0; i < 8; ++i) {
      const unsigned mo = (unsigned)(m0 + i + (half << 3));
      float v = acc[u][i] + bb[i];
      if (relu) v = fmaxf(v, 0.0f);
      Y[(bu[u] * (unsigned)M + mo) * strideT + tu[u]] = v;
    }
  }
}

// ---------------------------------------------------------------------------
// One tap of the k=3 conv: GEMM over ci at fixed dk. MASKED=false for the
// center tap (always in-bounds) removes the 0/1 mask multiplies.
// ---------------------------------------------------------------------------
template <int NT, bool MASKED>
__device__ __forceinline__ void
conv3_tap(const float* __restrict__ A, const float* __restrict__ X,
          unsigned offA, unsigned* offB, const float* msk,
          int Cin, unsigned strideT, v8f* acc)
{
  for (int ci = 0; ci < Cin; ci += 4) {
    __builtin_prefetch(&A[offA + 96], 0, 3);    // global_prefetch_b8 (near)
    const float a0 = A[offA];
    const float a1 = A[offA + 3];               // A stride 3 over ci
    offA += 12;
    float x0[NT], x1[NT];
#pragma unroll
    for (int u = 0; u < NT; ++u) {
      x0[u] = X[offB[u]];
      x1[u] = X[offB[u] + strideT];
      offB[u] += 4u * strideT;
    }
    const v2f av = { a0, a1 };
#pragma unroll
    for (int u = 0; u < NT; ++u) {
      v2f bv;
      if (MASKED) { bv.x = x0[u] * msk[u]; bv.y = x1[u] * msk[u]; }
      else        { bv.x = x0[u];          bv.y = x1[u]; }
      acc[u] = wmma4(av, bv, acc[u]);
    }
  }
}

// ---------------------------------------------------------------------------
// k=3 pad=1 conv as 3 shifted GEMMs (tap-major K order).
// A row-major (M, Cin*3): A[m][ci*3+dk]. Boundary handled by clamped offset
// + 0/1 float mask; EXEC is never perturbed (WMMA requires all-ones EXEC).
// ---------------------------------------------------------------------------
template <int NT>
__global__ void __launch_bounds__(32)
conv3_gemm_wmma(const float* __restrict__ A, const float* __restrict__ X,
                const float* __restrict__ bias, float* __restrict__ Y,
                int M, int Cin, int T, int relu)
{
  const int lane = threadIdx.x;
  const int row  = lane & 15;
  const int half = lane >> 4;
  const int n0 = blockIdx.x * (16 * NT);
  const int m0 = blockIdx.y << 4;
  const int mrow = m0 + row;
  const unsigned aRow = (unsigned)mrow * (unsigned)(Cin * 3);
  const unsigned strideT = (unsigned)T;

  int bu[NT], tu[NT];
  unsigned xb[NT];
#pragma unroll
  for (int u = 0; u < NT; ++u) {
    const int nc = n0 + 16 * u + row;
    const int b = nc / T;
    const int t = nc - b * T;
    bu[u] = b; tu[u] = t;
    xb[u] = (unsigned)b * Cin * T + (unsigned)(half << 1) * T;
  }

  v8f acc[NT];
#pragma unroll
  for (int u = 0; u < NT; ++u) acc[u] = {};

#pragma unroll
  for (int dk = 0; dk < 3; ++dk) {
    unsigned offB[NT];
    float msk[NT];
#pragma unroll
    for (int u = 0; u < NT; ++u) {
      const int tt = tu[u] + dk - 1;
      msk[u] = (tt >= 0 && tt < T) ? 1.0f : 0.0f;
      const int ttc = tt < 0 ? 0 : (tt >= T - 1 ? T - 1 : tt);
      offB[u] = xb[u] + (unsigned)ttc;
    }
    const unsigned offA = aRow + (unsigned)dk + (unsigned)(half << 1) * 3u;
    if (dk == 1)
      conv3_tap<NT, false>(A, X, offA, offB, msk, Cin, strideT, acc);
    else
      conv3_tap<NT, true >(A, X, offA, offB, msk, Cin, strideT, acc);
  }

  float bb[8];
#pragma unroll
  for (int i = 0; i < 8; ++i) bb[i] = bias[m0 + i + (half << 3)];
#pragma unroll
  for (int u = 0; u < NT; ++u) {
#pragma unroll
    for (int i = 0; i < 8; ++i) {
      const unsigned mo = (unsigned)(m0 + i + (half << 3));
      float v = acc[u][i] + bb[i];
      if (relu) v = fmaxf(v, 0.0f);
      Y[((unsigned)bu[u] * (unsigned)M + mo) * strideT + (unsigned)tu[u]] = v;
    }
  }
}

// ---------------------------------------------------------------------------
// Batched G[b][tde][ten] = sum_c Q[b][c][tde] * Kenc[b][c][ten]
// One wave computes 16 x (16*NT); grid.z = batch.
// ---------------------------------------------------------------------------
template <int NT>
__global__ void __launch_bounds__(32)
qk_gemm_wmma(const float* __restrict__ Q, const float* __restrict__ Kenc,
             float* __restrict__ G, int C, int Tde, int Ten)
{
  const int lane = threadIdx.x;
  const int row  = lane & 15;
  const int half = lane >> 4;
  const int n0 = blockIdx.x * (16 * NT);
  const int m0 = blockIdx.y << 4;
  const int b  = blockIdx.z;

  unsigned offQ = (unsigned)b * C * Tde + (unsigned)(half << 1) * Tde
                + (unsigned)(m0 + row);
  unsigned offK[NT];
#pragma unroll
  for (int u = 0; u < NT; ++u)
    offK[u] = (unsigned)b * C * Ten + (unsigned)(half << 1) * Ten
            + (unsigned)(n0 + 16 * u + row);

  v8f acc[NT];
#pragma unroll
  for (int u = 0; u < NT; ++u) acc[u] = {};

  for (int k = 0; k < C; k += 4) {
    const float a0 = Q[offQ];
    const float a1 = Q[offQ + (unsigned)Tde];
    offQ += 4u * (unsigned)Tde;
    float x0[NT], x1[NT];
#pragma unroll
    for (int u = 0; u < NT; ++u) {
      x0[u] = Kenc[offK[u]];
      x1[u] = Kenc[offK[u] + (unsigned)Ten];
      offK[u] += 4u * (unsigned)Ten;
    }
    const v2f av = { a0, a1 };
#pragma unroll
    for (int u = 0; u < NT; ++u) {
      const v2f bv = { x0[u], x1[u] };
      acc[u] = wmma4(av, bv, acc[u]);
    }
  }
#pragma unroll
  for (int u = 0; u < NT; ++u) {
#pragma unroll
    for (int i = 0; i < 8; ++i) {
      const unsigned mo = (unsigned)(m0 + i + (half << 3));
      G[((unsigned)b * Tde + mo) * (unsigned)Ten + (unsigned)(n0 + 16 * u + row)] =
          acc[u][i];
    }
  }
}

// out[b*T + t] = sum_c X[b][c][t]^2
__global__ void colsumsq_kernel(const float* __restrict__ X, float* __restrict__ out,
                                int C, int T, int total)
{
  const int n = blockIdx.x * blockDim.x + threadIdx.x;
  if (n >= total) return;
  const int b = n / T;
  const int t = n - b * T;
  unsigned off = (unsigned)b * C * T + (unsigned)t;
  float s = 0.0f;
  for (int c = 0; c < C; ++c) {
    const float v = X[off];
    s += v * v;
    off += (unsigned)T;
  }
  out[n] = s;
}

// One wave32 per (b, tde) row of Ten: logits -> log_softmax(+log prior) -> masked softmax
__global__ void __launch_bounds__(32)
attn_softmax_kernel(const float* __restrict__ G, const float* __restrict__ qq,
                    const float* __restrict__ kkn, const float* __restrict__ prior,
                    const unsigned char* __restrict__ mask,
                    float* __restrict__ attn, float* __restrict__ logprob,
                    int Tde, int Ten, float temp)
{
  const int rowid = blockIdx.x;       // b*Tde + tde
  const int b = rowid / Tde;
  const int lane = threadIdx.x;       // 0..31 (wave32)
  const int nIter = Ten >> 5;         // Ten multiple of 32 (160 -> 5)

  const float q2 = qq[rowid];
  const unsigned gbase = (unsigned)rowid * (unsigned)Ten;
  const unsigned kbase = (unsigned)b * (unsigned)Ten;

  float l[8];
  float mx = -INFINITY;
  for (int i = 0; i < nIter; ++i) {
    const int j = lane + (i << 5);
    const float v = -temp * (q2 + kkn[kbase + j] - 2.0f * G[gbase + j]);
    l[i] = v;
    mx = fmaxf(mx, v);
  }
  for (int off = 16; off > 0; off >>= 1) mx = fmaxf(mx, __shfl_xor(mx, off, 32));
  float s = 0.0f;
  for (int i = 0; i < nIter; ++i) s += expf(l[i] - mx);
  for (int off = 16; off > 0; off >>= 1) s += __shfl_xor(s, off, 32);
  const float lse = mx + logf(s);

  float mx2 = -INFINITY;
  for (int i = 0; i < nIter; ++i) {
    const int j = lane + (i << 5);
    const float lp = l[i] - lse + logf(prior[gbase + j] + 1e-8f);
    logprob[gbase + j] = lp;
    const float mv = mask[kbase + j] ? lp : -INFINITY;
    l[i] = mv;
    mx2 = fmaxf(mx2, mv);
  }
  for (int off = 16; off > 0; off >>= 1) mx2 = fmaxf(mx2, __shfl_xor(mx2, off, 32));
  float s2 = 0.0f;
  for (int i = 0; i < nIter; ++i) { l[i] = expf(l[i] - mx2); s2 += l[i]; }
  for (int off = 16; off > 0; off >>= 1) s2 += __shfl_xor(s2, off, 32);
  const float inv = 1.0f / s2;
  for (int i = 0; i < nIter; ++i) {
    const int j = lane + (i << 5);
    attn[gbase + j] = l[i] * inv;
  }
}

extern "C" void kernel_launch(void* const* d_in, const int* in_sizes, int n_in,
                              void* d_out, int out_size, void* d_ws, size_t ws_size,
                              hipStream_t stream) {
  (void)in_sizes; (void)n_in; (void)out_size; (void)ws_size;
  constexpr int B = 4, Cq = 80, Ck = 512, Ca = 80, Tde = 800, Ten = 160;
  constexpr float TEMP = 0.0005f;

  const float* queries = (const float*)d_in[0];
  const float* keys    = (const float*)d_in[1];
  const unsigned char* mask = (const unsigned char*)d_in[2];
  const float* prior   = (const float*)d_in[3];
  const float* wk1 = (const float*)d_in[4];  const float* bk1 = (const float*)d_in[5];
  const float* wk2 = (const float*)d_in[6];  const float* bk2 = (const float*)d_in[7];
  const float* wq1 = (const float*)d_in[8];  const float* bq1 = (const float*)d_in[9];
  const float* wq2 = (const float*)d_in[10]; const float* bq2 = (const float*)d_in[11];
  const float* wq3 = (const float*)d_in[12]; const float* bq3 = (const float*)d_in[13];

  float* ws = (float*)d_ws;
  size_t o = 0;
  float* y1   = ws + o; o += (size_t)B * 2 * Ck * Ten;   // key conv1 out (ReLU'd)
  float* q1   = ws + o; o += (size_t)B * 2 * Cq * Tde;   // query conv1 out
  float* kenc = ws + o; o += (size_t)B * Ca * Ten;
  float* q2b  = ws + o; o += (size_t)B * Cq * Tde;
  float* qenc = ws + o; o += (size_t)B * Ca * Tde;
  float* G    = ws + o; o += (size_t)B * Tde * Ten;
  float* qqv  = ws + o; o += (size_t)B * Tde;
  float* kkv  = ws + o; o += (size_t)B * Ten;

  const dim3 blk(32);

  // key_proj conv k=3 (512->1024)+ReLU : M=1024, N=640, K=1536 ; 64-wide strips
  conv3_gemm_wmma<4><<<dim3((B * Ten) / 64, (2 * Ck) / 16), blk, 0, stream>>>(
      wk1, keys, bk1, y1, 2 * Ck, Ck, Ten, 1);
  // key_proj 1x1 (1024->80) : M=80, N=640, K=1024
  conv1x1_gemm_wmma<4><<<dim3((B * Ten) / 64, Ca / 16), blk, 0, stream>>>(
      wk2, y1, bk2, kenc, Ca, 2 * Ck, Ten, 0);
  // query_proj conv k=3 (80->160)+ReLU : M=160, N=3200, K=240
  conv3_gemm_wmma<4><<<dim3((B * Tde) / 64, (2 * Cq) / 16), blk, 0, stream>>>(
      wq1, queries, bq1, q1, 2 * Cq, Cq, Tde, 1);
  // query_proj 1x1 (160->80)+ReLU
  conv1x1_gemm_wmma<4><<<dim3((B * Tde) / 64, Cq / 16), blk, 0, stream>>>(
      wq2, q1, bq2, q2b, Cq, 2 * Cq, Tde, 1);
  // query_proj 1x1 (80->80)
  conv1x1_gemm_wmma<4><<<dim3((B * Tde) / 64, Ca / 16), blk, 0, stream>>>(
      wq3, q2b, bq3, qenc, Ca, Cq, Tde, 0);

  // column squared norms
  colsumsq_kernel<<<(B * Tde + 255) / 256, 256, 0, stream>>>(qenc, qqv, Ca, Tde, B * Tde);
  colsumsq_kernel<<<(B * Ten + 255) / 256, 256, 0, stream>>>(kenc, kkv, Ca, Ten, B * Ten);

  // G = q_enc^T * k_enc per batch : M=800, N=160, K=80 ; 32-wide strips
  qk_gemm_wmma<2><<<dim3(Ten / 32, Tde / 16, B), blk, 0, stream>>>(
      qenc, kenc, G, Ca, Tde, Ten);

  // fused log_softmax + prior + masked softmax
  float* attn    = (float*)d_out;
  float* logprob = attn + (size_t)B * Tde * Ten;
  attn_softmax_kernel<<<B * Tde, blk, 0, stream>>>(G, qqv, kkv, prior, mask,
                                                   attn, logprob, Tde, Ten, TEMP);
}